// Stress_61074434949889
// MI455X (gfx1250) — compile-verified
//
#include <hip/hip_runtime.h>

typedef __attribute__((ext_vector_type(2))) float v2f;
typedef __attribute__((ext_vector_type(8))) float v8f;

#define N_ATOMS_C 2000000
#define N_STRUCT_C 16384

__device__ __forceinline__ float leakyf(float x) { return x >= 0.f ? x : 0.01f * x; }
__device__ __forceinline__ float reluf(float x)  { return x > 0.f ? x : 0.f; }

// ---------------------------------------------------------------------------
// Kernel 1: per-structure precompute.
//   cinv[s,9]   = inv(cell[s])                     (adjugate / det)
//   base32[s,n] = b2a[n] + sum_{i,j} cr_i cr_j * w2a[n, 9 + 9i + j]
//   out[s,0:6]  = 0
// ---------------------------------------------------------------------------
__global__ __launch_bounds__(256)
void stress_prep_kernel(const float* __restrict__ cell,
                        const float* __restrict__ w2a,
                        const float* __restrict__ b2a,
                        float* __restrict__ cinv,
                        float* __restrict__ base32,
                        float* __restrict__ out)
{
    int s = blockIdx.x * blockDim.x + threadIdx.x;
    if (s >= N_STRUCT_C) return;

    float m[9];
#pragma unroll
    for (int i = 0; i < 9; ++i) m[i] = cell[s * 9 + i];

    const float A = m[0], B = m[1], C = m[2];
    const float D = m[3], E = m[4], F = m[5];
    const float G = m[6], H = m[7], I = m[8];
    const float det = A * (E * I - F * H) - B * (D * I - F * G) + C * (D * H - E * G);
    const float r = 1.f / det;

    float inv[9];
    inv[0] = (E * I - F * H) * r; inv[1] = (C * H - B * I) * r; inv[2] = (B * F - C * E) * r;
    inv[3] = (F * G - D * I) * r; inv[4] = (A * I - C * G) * r; inv[5] = (C * D - A * F) * r;
    inv[6] = (D * H - E * G) * r; inv[7] = (B * G - A * H) * r; inv[8] = (A * E - B * D) * r;
#pragma unroll
    for (int i = 0; i < 9; ++i) cinv[s * 9 + i] = inv[i];

    for (int n = 0; n < 32; ++n) {
        float acc = b2a[n];
        const float* wrow = w2a + n * 90 + 9;
#pragma unroll
        for (int i = 0; i < 9; ++i) {
            const float ci = m[i];
            for (int j = 0; j < 9; ++j)
                acc += ci * m[j] * wrow[i * 9 + j];
        }
        base32[s * 32 + n] = acc;
    }

#pragma unroll
    for (int n = 0; n < 6; ++n) out[s * 6 + n] = 0.f;
}

// ---------------------------------------------------------------------------
// Kernel 2: per-atom MLP on the f32 WMMA path (16 atoms per wave).
//
// WMMA f32 16x16x4 fragment layouts (ISA 7.12.2):
//   A (16x4):  lane L holds row M = L%16;  v0 = K(2*(L/16)),   v1 = K+1
//   B (4x16):  lane L holds col N = L%16;  v0 = K(2*(L/16)),   v1 = K+1
//   C/D:       vgpr r, lane L -> M = r + 8*(L/16), N = L%16
//
// Weights staged zero-padded + transposed in LDS once per block (no EXEC-
// masked loads). batch is sorted -> tiles with first==last batch are
// single-structure: broadcast C-init and 6-atomics-per-tile reduction.
// ---------------------------------------------------------------------------
__global__ __launch_bounds__(256)
void stress_atoms_kernel(const float* __restrict__ ap,
                         const float* __restrict__ pos,
                         const int*   __restrict__ batch,
                         const float* __restrict__ w0,  const float* __restrict__ b0,
                         const float* __restrict__ w1,  const float* __restrict__ b1,
                         const float* __restrict__ w1n, const float* __restrict__ b1n,
                         const float* __restrict__ w2a,
                         const float* __restrict__ w2b, const float* __restrict__ b2b,
                         const float* __restrict__ w2c, const float* __restrict__ b2c,
                         const float* __restrict__ cinv,
                         const float* __restrict__ base32,
                         float* __restrict__ out)
{
    __shared__ float s_feat[8][16][12];   // 9 atom-varying features, padded to 12
    __shared__ int   s_batch[8][16];
    __shared__ float s_h[8][16 * 33];     // h32 transpose buffer (stride 33)
    __shared__ float s_h2[8][16 * 17];    // h16 transpose buffer (stride 17)
    __shared__ float wA[12][33];          // [K pad 12][N=32 pad 33] = w2a[:, :9].T padded
    __shared__ float wB[32][17];          // [K=32][N=16 pad 17]     = w2b.T
    __shared__ float wC[16][17];          // [K=16][N=16 pad 17]     = w2c.T, cols 6..15 = 0

    const int tid  = threadIdx.x;
    const int w    = tid >> 5;        // wave within block (wave32)
    const int lane = tid & 31;
    const int lo   = lane & 15;
    const int hi   = lane >> 4;
    const int tileBase = (blockIdx.x * 8 + w) * 16;   // 15625 * 8 * 16 == 2,000,000

    // ---- stage padded/transposed weights into LDS (whole block) -------------
    for (int idx = tid; idx < 12 * 32; idx += 256) {
        const int k = idx >> 5, n = idx & 31;
        wA[k][n] = (k < 9) ? w2a[n * 90 + k] : 0.f;
    }
    for (int idx = tid; idx < 32 * 16; idx += 256) {
        const int k = idx >> 4, n = idx & 15;
        wB[k][n] = w2b[n * 32 + k];
    }
    for (int idx = tid; idx < 16 * 16; idx += 256) {
        const int k = idx >> 4, n = idx & 15;
        wC[k][n] = (n < 6) ? w2c[n * 16 + k] : 0.f;
    }

    // ---- phase 1: lanes 0..15 build the 9 per-atom features -----------------
    if (lane < 16) {
        const int a = tileBase + lane;
        const int b = batch[a];
        s_batch[w][lane] = b;

        const float p0 = pos[a * 3], p1 = pos[a * 3 + 1], p2 = pos[a * 3 + 2];
        const float* ci = cinv + b * 9;

        float thr[3];
#pragma unroll
        for (int e = 0; e < 3; ++e) {
            const float fr = p0 * ci[e] + p1 * ci[3 + e] + p2 * ci[6 + e];
            thr[e] = fr - floorf(fr) - 0.5f;
        }

        const float a0 = ap[a * 3], a1 = ap[a * 3 + 1], a2 = ap[a * 3 + 2];
        float ap3[3], t1[3], t2[3];
#pragma unroll
        for (int i = 0; i < 3; ++i)
            ap3[i] = leakyf(a0 * w0[i * 3] + a1 * w0[i * 3 + 1] + a2 * w0[i * 3 + 2] + b0[i]);
#pragma unroll
        for (int i = 0; i < 3; ++i) {
            t1[i] = reluf( thr[0] * w1 [i * 3] + thr[1] * w1 [i * 3 + 1] + thr[2] * w1 [i * 3 + 2] + b1 [i]) * ap3[i];
            t2[i] = reluf(-thr[0] * w1n[i * 3] - thr[1] * w1n[i * 3 + 1] - thr[2] * w1n[i * 3 + 2] + b1n[i]) * ap3[i];
        }

        s_feat[w][lane][0] = a0;  s_feat[w][lane][1] = a1;  s_feat[w][lane][2] = a2;
#pragma unroll
        for (int i = 0; i < 3; ++i) {
            s_feat[w][lane][3 + i] = t1[i];
            s_feat[w][lane][6 + i] = t2[i];
            s_feat[w][lane][9 + i] = 0.f;
        }
    }
    __syncthreads();

    // batch is globally sorted -> first==last implies whole tile is one structure
    const int bFirst  = s_batch[w][0];
    const bool uniform = (bFirst == s_batch[w][15]);   // wave-uniform condition

    // ---- layer A: [16 x 12] @ [12 x 32], C init = per-structure base32 ------
    v8f c0, c1;
    if (uniform) {
        const float v0 = base32[bFirst * 32 + lo];
        const float v1 = base32[bFirst * 32 + 16 + lo];
#pragma unroll
        for (int r = 0; r < 8; ++r) { c0[r] = v0; c1[r] = v1; }
    } else {
#pragma unroll
        for (int r = 0; r < 8; ++r) {
            const int b = s_batch[w][r + 8 * hi];
            c0[r] = base32[b * 32 + lo];
            c1[r] = base32[b * 32 + 16 + lo];
        }
    }
#pragma unroll
    for (int k = 0; k < 3; ++k) {
        const int k0 = 4 * k + 2 * hi, k1 = k0 + 1;
        v2f av, bv0, bv1;
        av.x  = s_feat[w][lo][k0];
        av.y  = s_feat[w][lo][k1];
        bv0.x = wA[k0][lo];      bv0.y = wA[k1][lo];
        bv1.x = wA[k0][16 + lo]; bv1.y = wA[k1][16 + lo];
        c0 = __builtin_amdgcn_wmma_f32_16x16x4_f32(false, av, false, bv0, (short)0, c0, false, false);
        c1 = __builtin_amdgcn_wmma_f32_16x16x4_f32(false, av, false, bv1, (short)0, c1, false, false);
    }
#pragma unroll
    for (int r = 0; r < 8; ++r) {
        const int m = r + 8 * hi;
        s_h[w][m * 33 + lo]      = reluf(c0[r]);
        s_h[w][m * 33 + 16 + lo] = reluf(c1[r]);
    }
    __syncthreads();

    // ---- layer B: [16 x 32] @ [32 x 16] -------------------------------------
    v8f cB;
    {
        const float bb = b2b[lo];
#pragma unroll
        for (int r = 0; r < 8; ++r) cB[r] = bb;
    }
#pragma unroll
    for (int k = 0; k < 8; ++k) {
        const int k0 = 4 * k + 2 * hi;
        v2f av, bv;
        av.x = s_h[w][lo * 33 + k0];
        av.y = s_h[w][lo * 33 + k0 + 1];
        bv.x = wB[k0][lo];
        bv.y = wB[k0 + 1][lo];
        cB = __builtin_amdgcn_wmma_f32_16x16x4_f32(false, av, false, bv, (short)0, cB, false, false);
    }
#pragma unroll
    for (int r = 0; r < 8; ++r)
        s_h2[w][(r + 8 * hi) * 17 + lo] = leakyf(cB[r]);
    __syncthreads();

    // ---- layer C: [16 x 16] @ [16 x 16] (cols 6..15 zero-padded) ------------
    v8f cC;
    {
        const float bc = (lo < 6) ? b2c[lo] : 0.f;
#pragma unroll
        for (int r = 0; r < 8; ++r) cC[r] = bc;
    }
#pragma unroll
    for (int k = 0; k < 4; ++k) {
        const int k0 = 4 * k + 2 * hi;
        v2f av, bv;
        av.x = s_h2[w][lo * 17 + k0];
        av.y = s_h2[w][lo * 17 + k0 + 1];
        bv.x = wC[k0][lo];
        bv.y = wC[k0 + 1][lo];
        cC = __builtin_amdgcn_wmma_f32_16x16x4_f32(false, av, false, bv, (short)0, cC, false, false);
    }

    // ---- segment-sum scatter -------------------------------------------------
    if (uniform) {
        // whole tile -> one structure: reduce 16 atoms in-wave, 6 atomics total
        float partial = 0.f;
#pragma unroll
        for (int r = 0; r < 8; ++r) partial += cC[r];
        partial += __shfl_xor(partial, 16, 32);   // combine atom halves (lane L <-> L+16)
        if (hi == 0 && lo < 6)
            atomicAdd(&out[bFirst * 6 + lo], partial);
    } else {
        if (lo < 6) {
#pragma unroll
            for (int r = 0; r < 8; ++r) {
                const int b = s_batch[w][r + 8 * hi];
                atomicAdd(&out[b * 6 + lo], cC[r]);
            }
        }
    }
}

// ---------------------------------------------------------------------------
// Harness entry point.
// d_in order: atom_prop, pos, cell, batch, w0, b0, w1, b1, w1n, b1n,
//             w2a, b2a, w2b, b2b, w2c, b2c
// ---------------------------------------------------------------------------
extern "C" void kernel_launch(void* const* d_in, const int* in_sizes, int n_in,
                              void* d_out, int out_size, void* d_ws, size_t ws_size,
                              hipStream_t stream)
{
    const float* atom_prop = (const float*)d_in[0];
    const float* pos       = (const float*)d_in[1];
    const float* cell      = (const float*)d_in[2];
    const int*   batch     = (const int*)  d_in[3];
    const float* w0  = (const float*)d_in[4];
    const float* b0  = (const float*)d_in[5];
    const float* w1  = (const float*)d_in[6];
    const float* b1  = (const float*)d_in[7];
    const float* w1n = (const float*)d_in[8];
    const float* b1n = (const float*)d_in[9];
    const float* w2a = (const float*)d_in[10];
    const float* b2a = (const float*)d_in[11];
    const float* w2b = (const float*)d_in[12];
    const float* b2b = (const float*)d_in[13];
    const float* w2c = (const float*)d_in[14];
    const float* b2c = (const float*)d_in[15];
    float* out = (float*)d_out;

    // workspace: cinv [S,9] then base32 [S,32]  (~2.6 MB)
    float* ws_cinv   = (float*)d_ws;
    float* ws_base32 = ws_cinv + (size_t)N_STRUCT_C * 9;

    stress_prep_kernel<<<N_STRUCT_C / 256, 256, 0, stream>>>(
        cell, w2a, b2a, ws_cinv, ws_base32, out);

    // 2,000,000 atoms = 15625 blocks * 8 waves * 16 atoms (exact, no tail)
    stress_atoms_kernel<<<N_ATOMS_C / 128, 256, 0, stream>>>(
        atom_prop, pos, batch, w0, b0, w1, b1, w1n, b1n,
        w2a, w2b, b2b, w2c, b2c, ws_cinv, ws_base32, out);
}